// QMIXNetwork_68126771249417
// MI455X (gfx1250) — compile-verified
//
#include <hip/hip_runtime.h>

typedef __attribute__((ext_vector_type(2))) float v2f;
typedef __attribute__((ext_vector_type(4))) float v4f;
typedef __attribute__((ext_vector_type(8))) float v8f;

#define QMIX_AGENTS 8
#define QMIX_K      256     // STATE_DIM
#define QMIX_NTOT   352     // 256 (w1) + 32 (wf) + 32 (b1) + 32 (v1)
#define QMIX_NTILES 22      // 352 / 16
#define QMIX_FPT    34      // fragments per tile: 32 W + 2 bias
#define QMIX_NFRAG  (QMIX_NTILES * QMIX_FPT)   // 748
#define QMIX_BATCH  65536
#define QMIX_PF     6       // fragment prefetch depth

// Tile visit order: wf/b1/V-head first (so b1 seeds hacc, V head folds to a
// scalar), then the 16 w1 tiles.
__host__ __device__ __forceinline__ constexpr int qmix_tile(int idx) {
  return (idx < 6) ? (16 + idx) : (idx - 6);
}

// ---------------------------------------------------------------------------
// Repack kernel: build the per-tile fragment stream in workspace.
// Fragment f = t*34 + j, flat layout Wp[f*128 + lane*4 + d].
//  j < 32  : WMMA A-fragment of W^T tile t, k-step pair j:
//              n (output col) = t*16 + (lane&15)
//              physical k     = 8*j + 4*(lane>>4) + 2*(d>>1) + (d&1)
//            (K-permutation makes the matching X^T B-fragment one contiguous
//             float4 per lane: X[row][8*j + 4*hi .. +3])
//  j = 32/33: lane-replicated bias fragment: bias[t-region][hi*8 + (j-32)*4 + d]
// ---------------------------------------------------------------------------
__global__ __launch_bounds__(256) void qmix_repack(
    const float* __restrict__ w1W, const float* __restrict__ wfW,
    const float* __restrict__ b1W, const float* __restrict__ v1W,
    const float* __restrict__ w1b, const float* __restrict__ wfb,
    const float* __restrict__ b1b, const float* __restrict__ v1b,
    float* __restrict__ Wp) {
  int tid = blockIdx.x * 256 + threadIdx.x;
  if (tid >= QMIX_NFRAG * 32) return;
  int f    = tid >> 5;        // fragment (0..747)
  int lane = tid & 31;
  int t    = f / QMIX_FPT;    // tile (0..21)
  int j    = f % QMIX_FPT;    // fragment-in-tile (0..33)
  int hi   = lane >> 4;
  v4f o;
  if (j < 32) {
    int n = t * 16 + (lane & 15);
#pragma unroll
    for (int d = 0; d < 4; ++d) {
      int k = 8 * j + 4 * hi + 2 * (d >> 1) + (d & 1);
      float v;
      if (n < 256)      v = w1W[k * 256 + n];
      else if (n < 288) v = wfW[k * 32 + (n - 256)];
      else if (n < 320) v = b1W[k * 32 + (n - 288)];
      else              v = v1W[k * 32 + (n - 320)];
      o[d] = v;
    }
  } else {
    const float* bsrc;
    int boff;
    if (t < 16)      { bsrc = w1b; boff = t * 16; }
    else if (t < 18) { bsrc = wfb; boff = (t - 16) * 16; }
    else if (t < 20) { bsrc = b1b; boff = (t - 18) * 16; }
    else             { bsrc = v1b; boff = (t - 20) * 16; }
#pragma unroll
    for (int d = 0; d < 4; ++d)
      o[d] = bsrc[boff + hi * 8 + (j - 32) * 4 + d];
  }
  *(v4f*)(Wp + (size_t)tid * 4) = o;
}

// ---------------------------------------------------------------------------
// Main kernel: one wave computes 16 batch rows via D = W^T (352x256) x X^T
// (256x16), 22 tiles x 64 k-steps of V_WMMA_F32_16X16X4_F32.
// All per-tile operands (W fragments + bias fragments) flow through one
// register prefetch queue (~6 b128 loads in flight). Epilogue is lane-local;
// one shfl_xor(16) combines the two lane halves per batch row.
// ---------------------------------------------------------------------------
__global__ __launch_bounds__(256)
__attribute__((amdgpu_num_vgpr(256), amdgpu_waves_per_eu(4)))
void qmix_main(
    const float* __restrict__ Q,    // agent_qs [B,8]
    const float* __restrict__ X,    // states   [B,256]
    const float* __restrict__ v2W, const float* __restrict__ v2b,
    const float* __restrict__ Wp,   // repacked fragment stream
    float* __restrict__ out) {
  const int lane = threadIdx.x & 31;
  const int wave = threadIdx.x >> 5;
  const int hi   = lane >> 4;
  const int bcol = lane & 15;
  const int row  = blockIdx.x * 128 + wave * 16 + bcol;

  // Preload all B fragments (X^T): pair p covers k-steps 2p, 2p+1.
  v4f bpair[32];
  const float* xrow = X + (size_t)row * QMIX_K + hi * 4;
#pragma unroll
  for (int p = 0; p < 32; ++p)
    bpair[p] = __builtin_nontemporal_load((const v4f*)(xrow + 8 * p));

  // Lane-local epilogue registers; index j -> e = (j>>3)*16 + (j&7) + 8*hi
  float hacc[16];   // b1 (with bias), then += q[a]*|y1|
  float wfv[16];    // |states@wfW + wfb|
  float sv = 0.0f;  // V-head partial (relu(v1) . v2_W)

  // Fragment prefetch queue: fragment g of the visit sequence lives in
  // wq[g % QMIX_PF]; consuming step g reloads the slot with fragment g+PF.
  const float* wlane = Wp + lane * 4;
  v4f wq[QMIX_PF];
#pragma unroll
  for (int i = 0; i < QMIX_PF; ++i)   // idx 0 -> tile 16, fragment i
    wq[i] = *(const v4f*)(wlane + (size_t)(16 * QMIX_FPT + i) * 128);

#pragma unroll
  for (int idx = 0; idx < QMIX_NTILES; ++idx) {
    const int t = qmix_tile(idx);

    v8f c = {0.f, 0.f, 0.f, 0.f, 0.f, 0.f, 0.f, 0.f};
    v4f bv0, bv1;

#pragma unroll
    for (int j = 0; j < QMIX_FPT; ++j) {
      const int g = idx * QMIX_FPT + j;
      const v4f cur = wq[g % QMIX_PF];
      const int gn = g + QMIX_PF;
      if (gn < QMIX_NFRAG) {
        const int tn = qmix_tile(gn / QMIX_FPT);
        const int jn = gn % QMIX_FPT;
        wq[g % QMIX_PF] = *(const v4f*)(wlane + (size_t)(tn * QMIX_FPT + jn) * 128);
      }
      if (j < 32) {
        v2f a0 = {cur[0], cur[1]};
        v2f a1 = {cur[2], cur[3]};
        v2f b0 = {bpair[j][0], bpair[j][1]};
        v2f b1 = {bpair[j][2], bpair[j][3]};
        c = __builtin_amdgcn_wmma_f32_16x16x4_f32(false, a0, false, b0,
                                                  (short)0, c, false, false);
        c = __builtin_amdgcn_wmma_f32_16x16x4_f32(false, a1, false, b1,
                                                  (short)0, c, false, false);
      } else if (j == 32) {
        bv0 = cur;
      } else {
        bv1 = cur;
      }
    }

    const int p8 = (t & 1) * 8;
    if (t < 16) {            // y1 region: hacc[j] += q[a] * |y1 + b|
      const float qa = Q[(size_t)row * 8 + (t >> 1)];
#pragma unroll
      for (int r = 0; r < 8; ++r) {
        const float b = (r < 4) ? bv0[r & 3] : bv1[r & 3];
        hacc[p8 + r] += qa * __builtin_fabsf(c[r] + b);
      }
    } else if (t < 18) {     // w_final = |states@wfW + wfb|
#pragma unroll
      for (int r = 0; r < 8; ++r) {
        const float b = (r < 4) ? bv0[r & 3] : bv1[r & 3];
        wfv[p8 + r] = __builtin_fabsf(c[r] + b);
      }
    } else if (t < 20) {     // b1 seeds hacc
#pragma unroll
      for (int r = 0; r < 8; ++r) {
        const float b = (r < 4) ? bv0[r & 3] : bv1[r & 3];
        hacc[p8 + r] = c[r] + b;
      }
    } else {                 // V head: fold relu(.)*v2_W immediately
      const v4f va = *(const v4f*)(v2W + (t - 20) * 16 + hi * 8);
      const v4f vb = *(const v4f*)(v2W + (t - 20) * 16 + hi * 8 + 4);
#pragma unroll
      for (int r = 0; r < 8; ++r) {
        const float b = (r < 4) ? bv0[r & 3] : bv1[r & 3];
        float rv = c[r] + b;
        rv = (rv > 0.0f) ? rv : 0.0f;
        sv += rv * ((r < 4) ? va[r & 3] : vb[r & 3]);
      }
    }
  }

  float tot = sv;
#pragma unroll
  for (int j = 0; j < 16; ++j) {
    float h = hacc[j];                       // agent_qs . w1 + b1
    h = (h > 0.0f) ? h : (__expf(h) - 1.0f); // elu
    tot += h * wfv[j];                       // . w_final
  }
  tot += __shfl_xor(tot, 16, 32);            // combine lane halves
  if (hi == 0)
    __builtin_nontemporal_store(tot + v2b[0], out + row);
}

// ---------------------------------------------------------------------------
extern "C" void kernel_launch(void* const* d_in, const int* in_sizes, int n_in,
                              void* d_out, int out_size, void* d_ws, size_t ws_size,
                              hipStream_t stream) {
  (void)in_sizes; (void)n_in; (void)out_size; (void)ws_size;
  const float* q   = (const float*)d_in[0];
  const float* X   = (const float*)d_in[1];
  const float* w1W = (const float*)d_in[2];
  const float* w1b = (const float*)d_in[3];
  const float* wfW = (const float*)d_in[4];
  const float* wfb = (const float*)d_in[5];
  const float* b1W = (const float*)d_in[6];
  const float* b1b = (const float*)d_in[7];
  const float* v1W = (const float*)d_in[8];
  const float* v1b = (const float*)d_in[9];
  const float* v2W = (const float*)d_in[10];
  const float* v2b = (const float*)d_in[11];
  float* Wp  = (float*)d_ws;                  // 748*32*4 floats ~= 374 KiB
  float* out = (float*)d_out;

  qmix_repack<<<(QMIX_NFRAG * 32 + 255) / 256, 256, 0, stream>>>(
      w1W, wfW, b1W, v1W, w1b, wfb, b1b, v1b, Wp);
  qmix_main<<<QMIX_BATCH / 128, 256, 0, stream>>>(q, X, v2W, v2b, Wp, out);
}